// AudioModel_86689619902579
// MI455X (gfx1250) — compile-verified
//
#include <hip/hip_runtime.h>

typedef __attribute__((ext_vector_type(2))) float v2f;
typedef __attribute__((ext_vector_type(4))) float v4f;
typedef __attribute__((ext_vector_type(8))) float v8f;

#define B_DIM 128
#define T_DIM 1496
#define C_DIM 768
#define ROWS (B_DIM * T_DIM)          // 191488, divisible by 16
#define WAVES_PER_BLOCK 8
#define TILE_ROWS 16

// ---------------------------------------------------------------------------
// Phase A: d[row] = h[row,:] . W  via V_WMMA_F32_16X16X4_F32.
// One wave32 per 16-row tile. Per 8-K macro chunk each lane does ONE
// global_load_b128 of h (non-temporal, streamed once) and one ds_load_b128
// of the LDS-staged W with the *same* lane/VGPR placement. Because the
// K-slot permutation is applied identically to A and B, the contraction
// pairs each h value with its matching W value exactly. Two WMMAs consume
// VGPR pairs (0,1) and (2,3). All 16 D columns are identical; lane0 holds
// d[rows 0..7] (M=0..7) and lane16 holds d[rows 8..15].
// ---------------------------------------------------------------------------
__global__ __launch_bounds__(256) void audio_dot_wmma(
    const float* __restrict__ h, const float* __restrict__ W,
    float* __restrict__ d) {
  __shared__ float sW[C_DIM];
  for (int i = threadIdx.x; i < C_DIM; i += 256) sW[i] = W[i];
  __syncthreads();

  const int lane = threadIdx.x & 31;
  const int wave = blockIdx.x * WAVES_PER_BLOCK + (threadIdx.x >> 5);
  const int half = lane >> 4;     // 0: lanes 0-15, 1: lanes 16-31
  const int m    = lane & 15;     // row within tile

  const long rowBase = (long)wave * TILE_ROWS;
  const float* __restrict__ hbase = h + (rowBase + m) * C_DIM + half * 4;
  const float* wbase = sW + half * 4;

  v8f c = {};
#pragma unroll 4
  for (int k0 = 0; k0 < C_DIM; k0 += 8) {
    // streamed tensor: one b128 non-temporal load per lane per 8 K-values
    v4f a = __builtin_nontemporal_load((const v4f*)(hbase + k0));
    // W chunk from LDS (DS pipe, immediate offsets after unroll)
    v4f w = *(const v4f*)(wbase + k0);
    v2f a01 = {a.x, a.y}, a23 = {a.z, a.w};
    v2f w01 = {w.x, w.y}, w23 = {w.z, w.w};
    // 8 args: (neg_a, A, neg_b, B, c_mod, C, reuse_a, reuse_b)
    c = __builtin_amdgcn_wmma_f32_16x16x4_f32(
        false, a01, false, w01, (short)0, c, false, false);
    c = __builtin_amdgcn_wmma_f32_16x16x4_f32(
        false, a23, false, w23, (short)0, c, false, false);
  }

  // D layout: VGPR i, lanes 0-15 -> M=i (N=lane); lanes 16-31 -> M=8+i.
  if (lane == 0) {
#pragma unroll
    for (int i = 0; i < 8; ++i) d[rowBase + i] = c[i];
  } else if (lane == 16) {
#pragma unroll
    for (int i = 0; i < 8; ++i) d[rowBase + 8 + i] = c[i];
  }
}

// ---------------------------------------------------------------------------
// Phase B: one block per batch row. Segment the phoneme runs, ragged segment
// mean of d over each run, mean over runs, add bias.
// ---------------------------------------------------------------------------
#define NT 256
#define ITEMS ((T_DIM + NT - 1) / NT)   // 6

__global__ __launch_bounds__(NT) void audio_run_reduce(
    const float* __restrict__ d, const int* __restrict__ pid,
    const int* __restrict__ lens, const float* __restrict__ bias,
    float* __restrict__ out) {
  __shared__ float segsum[T_DIM];
  __shared__ int   segcnt[T_DIM];
  __shared__ int   tbase[NT];
  __shared__ float wsum[NT / 32];
  __shared__ int   s_nruns;

  const int b  = blockIdx.x;
  const int th = threadIdx.x;
  const int* __restrict__ prow = pid + (long)b * T_DIM;
  const float* __restrict__ drow = d + (long)b * T_DIM;

  int len = lens[b];
  if (len < 1) len = 1;
  if (len > T_DIM) len = T_DIM;

  for (int i = th; i < T_DIM; i += NT) { segsum[i] = 0.f; segcnt[i] = 0; }

  // boundary flags for this thread's contiguous chunk
  const int t0 = th * ITEMS;
  int flags[ITEMS];
  int total = 0;
#pragma unroll
  for (int i = 0; i < ITEMS; ++i) {
    const int t = t0 + i;
    int f = 0;
    if (t < len) f = (t == 0) ? 1 : (prow[t] != prow[t - 1]);
    flags[i] = f;
    total += f;
  }
  tbase[th] = total;
  __syncthreads();
  if (th == 0) {               // exclusive scan of 256 per-thread totals
    int run = 0;
    for (int i = 0; i < NT; ++i) { int v = tbase[i]; tbase[i] = run; run += v; }
    s_nruns = run;
  }
  __syncthreads();

  // scatter-add per-run sums/counts into LDS
  int s = tbase[th];
#pragma unroll
  for (int i = 0; i < ITEMS; ++i) {
    const int t = t0 + i;
    if (t < len) {
      s += flags[i];
      const int seg = s - 1;
      atomicAdd(&segsum[seg], drow[t]);
      atomicAdd(&segcnt[seg], 1);
    }
  }
  __syncthreads();

  const int nruns = s_nruns;
  float acc = 0.f;
  for (int sg = th; sg < nruns; sg += NT)
    acc += segsum[sg] / (float)segcnt[sg];

  // wave32 reduction + cross-wave via LDS
  for (int off = 16; off > 0; off >>= 1) acc += __shfl_down(acc, off, 32);
  if ((th & 31) == 0) wsum[th >> 5] = acc;
  __syncthreads();
  if (th == 0) {
    float tot = 0.f;
#pragma unroll
    for (int i = 0; i < NT / 32; ++i) tot += wsum[i];
    out[b] = tot / (float)nruns + bias[0];
  }
}

// ---------------------------------------------------------------------------
extern "C" void kernel_launch(void* const* d_in, const int* in_sizes, int n_in,
                              void* d_out, int out_size, void* d_ws,
                              size_t ws_size, hipStream_t stream) {
  const float* h    = (const float*)d_in[0];   // (B,T,C) f32
  const float* W    = (const float*)d_in[1];   // (1,C)   f32
  const float* bias = (const float*)d_in[2];   // (1,)    f32
  const int*   pid  = (const int*)d_in[3];     // (B,T)   i32
  const int*   lens = (const int*)d_in[4];     // (B,)    i32
  float* out = (float*)d_out;                  // (B,1)   f32
  float* dvec = (float*)d_ws;                  // B*T f32 scratch (~766 KB)

  const int tiles = ROWS / TILE_ROWS;                  // 11968 waves
  const int blocksA = tiles / WAVES_PER_BLOCK;         // 1496 blocks
  audio_dot_wmma<<<blocksA, 256, 0, stream>>>(h, W, dvec);
  audio_run_reduce<<<B_DIM, NT, 0, stream>>>(dvec, pid, lens, bias, out);
}